// LearnableTD_79688823210636
// MI455X (gfx1250) — compile-verified
//
#include <hip/hip_runtime.h>
#include <hip/hip_bf16.h>
#include <stdint.h>

// ---------------------------------------------------------------------------
// LearnableTD lambda-return backward scan, MI455X (gfx1250).
//
// Memory-bound (~134 MB @ 23.3 TB/s => ~5.8us floor). Strategy:
//   * wave-cooperative, fully coalesced tile loads staged through LDS using
//     gfx1250 async global->LDS ops (ASYNCcnt), double-buffered
//   * per-lane sequential recurrence consumed from LDS (ds_load_b128,
//     conflict-free via 20-dword padded row stride)
//   * uniform per-step coefficients p=gamma*lam, q=gamma*(1-lam) in LDS
// ---------------------------------------------------------------------------

#define EPSF 1e-8f

constexpr int S_LEN = 1024;     // sequence length
constexpr int SV    = 1025;     // values row stride (S+1)
constexpr int TILE  = 16;       // t-steps per tile
constexpr int NT    = S_LEN / TILE;   // 64 tiles
constexpr int RSTR  = 20;       // LDS row stride (dwords): 80B -> 16B aligned,
                                // and lane*20 mod 64 banks => only 2-way conflicts
constexpr int WAVES = 8;
constexpr int BLOCK = WAVES * 32;
constexpr unsigned BUF_BYTES  = 32u * RSTR * 4u;   // 2560 B per buffer per array
constexpr unsigned WAVE_BYTES = 6u * BUF_BYTES;    // r/d/v double-buffered: 15360 B
constexpr unsigned PQ_BYTES   = 2u * S_LEN * 4u;   // 8192 B
constexpr unsigned SMEM_BYTES = PQ_BYTES + WAVES * WAVE_BYTES; // 131072 B

// ---- gfx1250 async global->LDS ops (tracked by ASYNCcnt) -------------------
__device__ __forceinline__ void async_b128(unsigned lds_byte, const void* gptr) {
  asm volatile("global_load_async_to_lds_b128 %0, %1, off"
               :: "v"(lds_byte), "v"(gptr) : "memory");
}
__device__ __forceinline__ void async_b32(unsigned lds_byte, const void* gptr) {
  asm volatile("global_load_async_to_lds_b32 %0, %1, off"
               :: "v"(lds_byte), "v"(gptr) : "memory");
}
template <int N>
__device__ __forceinline__ void wait_async() {
  asm volatile("s_wait_asynccnt %0" :: "n"(N) : "memory");
}

// ---------------------------------------------------------------------------
// Kernel 1: sum_reward_weights (tiny). Parallelized via 32-chunk linear
// recurrence decomposition: v_t = q_t + p_t * v_{t+1}, v_S = 1,
// w_t = max(1 - v_t, eps), srw = w / max(mean(w), eps).
// ---------------------------------------------------------------------------
__global__ void __launch_bounds__(1024) td_weights_kernel(
    const float* __restrict__ raw_gamma, const float* __restrict__ raw_lambd,
    float* __restrict__ srw_out)
{
  __shared__ float shP[S_LEN];
  __shared__ float shQ[S_LEN];
  __shared__ float shW[S_LEN];
  __shared__ float shA[32], shB[32], shSeed[32];
  __shared__ float red[1024];

  const int tid = threadIdx.x;
  const float gamma = fmaxf(tanhf(raw_gamma[0]), EPSF);
  const float lam   = fmaxf(tanhf(raw_lambd[tid]), EPSF);
  shP[tid] = gamma * lam;
  shQ[tid] = gamma * (1.0f - lam);
  __syncthreads();

  // per-chunk composition: v_start = B + A * v_end over 32 steps
  if (tid < 32) {
    float A = 1.0f, Bc = 0.0f;
    for (int k = 31; k >= 0; --k) {
      const int t = tid * 32 + k;
      Bc = fmaf(shP[t], Bc, shQ[t]);
      A  = shP[t] * A;
    }
    shA[tid] = A; shB[tid] = Bc;
  }
  __syncthreads();
  if (tid == 0) {               // serial 32-step chunk-carry combine
    float x = 1.0f;             // v at t = S
    for (int c = 31; c >= 0; --c) { shSeed[c] = x; x = fmaf(shA[c], x, shB[c]); }
  }
  __syncthreads();
  if (tid < 32) {               // recompute chunk with correct seed, emit w
    float v = shSeed[tid];
    for (int k = 31; k >= 0; --k) {
      const int t = tid * 32 + k;
      v = fmaf(shP[t], v, shQ[t]);
      shW[t] = fmaxf(1.0f - v, EPSF);
    }
  }
  __syncthreads();

  red[tid] = shW[tid];
  __syncthreads();
  for (int s = 512; s > 0; s >>= 1) {
    if (tid < s) red[tid] += red[tid + s];
    __syncthreads();
  }
  const float mean  = red[0] * (1.0f / (float)S_LEN);
  const float scale = 1.0f / fmaxf(mean, EPSF);
  srw_out[tid] = shW[tid] * scale;
}

// ---------------------------------------------------------------------------
// Kernel 2: main backward scan. 8 waves/block, 32 rows/wave.
// Per tile (32 rows x 16 steps) a wave issues 24 async ops:
//   rewards: 4 x b128 (16B/lane, coalesced, 16B-aligned)
//   dones  : 4 x b128
//   values : 16 x b32  (row stride 1025 dwords => 4B phase, b32 only)
// Double buffered; wait_asynccnt<=24 guarantees the current tile landed
// (async loads complete in order).
// ---------------------------------------------------------------------------
__global__ void __launch_bounds__(BLOCK) td_scan_kernel(
    const float* __restrict__ raw_gamma, const float* __restrict__ raw_lambd,
    const float* __restrict__ values,  const float* __restrict__ rewards,
    const int*   __restrict__ dones,   float* __restrict__ out)
{
  extern __shared__ __align__(16) char smem[];
  float* pqf = (float*)smem;                 // interleaved [p,q] per t : 8KB

  const int tid  = threadIdx.x;
  const int lane = tid & 31;
  const int wave = tid >> 5;

  // recompute p/q per block (1024 tanhf per block: negligible)
  {
    const float gamma = fmaxf(tanhf(raw_gamma[0]), EPSF);
    for (int i = tid; i < S_LEN; i += BLOCK) {
      const float lam = fmaxf(tanhf(raw_lambd[i]), EPSF);
      pqf[2*i]   = gamma * lam;          // p
      pqf[2*i+1] = gamma * (1.0f - lam); // q
    }
  }
  __syncthreads();

  // LDS byte offsets for async destinations: dynamic shared starts at
  // groupstaticsize() (== 0: no static LDS in this kernel).
  const unsigned lds0  = (unsigned)__builtin_amdgcn_groupstaticsize();
  const unsigned wbase = lds0 + PQ_BYTES + (unsigned)wave * WAVE_BYTES;
  char* wptr = smem + PQ_BYTES + wave * WAVE_BYTES;   // generic view for reads

  const int R0 = (blockIdx.x * WAVES + wave) * 32;    // wave's 32 rows

  // per-lane constant pieces of the cooperative load pattern
  const unsigned lds_rd = (((lane >> 2) * RSTR) + ((lane & 3) << 2)) * 4u;
  const unsigned lds_v  = (((lane >> 4) * RSTR) + (lane & 15)) * 4u;
  const float* rL = rewards + (size_t)(R0 + (lane >> 2)) * S_LEN + ((lane & 3) << 2);
  const int*   dL = dones   + (size_t)(R0 + (lane >> 2)) * S_LEN + ((lane & 3) << 2);
  const float* vL = values  + (size_t)(R0 + (lane >> 4)) * SV + (lane & 15) + 1;

  float  ret    = values[(size_t)(R0 + lane) * SV + S_LEN]; // values[:, -1]
  float* outRow = out + (size_t)(R0 + lane) * S_LEN;

  auto issue = [&](int k) {
    const int t0 = k * TILE;
    const unsigned b = (unsigned)(k & 1);
    const unsigned ldsRb = wbase + b * BUF_BYTES;
    const unsigned ldsDb = wbase + 2u * BUF_BYTES + b * BUF_BYTES;
    const unsigned ldsVb = wbase + 4u * BUF_BYTES + b * BUF_BYTES;
#pragma unroll
    for (int i = 0; i < 4; ++i) {   // 8 rows x 16 floats each, coalesced
      async_b128(ldsRb + lds_rd + (unsigned)(i * 8 * RSTR * 4), rL + t0 + i * 8 * S_LEN);
      async_b128(ldsDb + lds_rd + (unsigned)(i * 8 * RSTR * 4), dL + t0 + i * 8 * S_LEN);
    }
#pragma unroll
    for (int j = 0; j < 16; ++j) {  // 2 rows x 16 dwords each
      async_b32(ldsVb + lds_v + (unsigned)(j * 2 * RSTR * 4), vL + t0 + j * 2 * SV);
    }
  };

  issue(NT - 1);                                  // prologue
  for (int k = NT - 1; k >= 0; --k) {
    if (k > 0) { issue(k - 1); wait_async<24>(); }   // overlap next tile
    else       { wait_async<0>(); }
    const unsigned b = (unsigned)(k & 1);
    const float* ldsRw = (const float*)(wptr + b * BUF_BYTES);
    const int*   ldsDw = (const int*  )(wptr + 2u * BUF_BYTES + b * BUF_BYTES);
    const float* ldsVw = (const float*)(wptr + 4u * BUF_BYTES + b * BUF_BYTES);
    const int t0 = k * TILE;
#pragma unroll
    for (int g = 3; g >= 0; --g) {
      const float4 r4 = *(const float4*)(ldsRw + lane * RSTR + g * 4);
      const int4   d4 = *(const int4*  )(ldsDw + lane * RSTR + g * 4);
      const float4 v4 = *(const float4*)(ldsVw + lane * RSTR + g * 4);
      float4 o;
      { const float2 pq = *(const float2*)(pqf + 2 * (t0 + g * 4 + 3));
        const float c = fmaf(pq.x, ret, pq.y * v4.w);
        ret = r4.w + (d4.w ? 0.0f : c); o.w = ret; }
      { const float2 pq = *(const float2*)(pqf + 2 * (t0 + g * 4 + 2));
        const float c = fmaf(pq.x, ret, pq.y * v4.z);
        ret = r4.z + (d4.z ? 0.0f : c); o.z = ret; }
      { const float2 pq = *(const float2*)(pqf + 2 * (t0 + g * 4 + 1));
        const float c = fmaf(pq.x, ret, pq.y * v4.y);
        ret = r4.y + (d4.y ? 0.0f : c); o.y = ret; }
      { const float2 pq = *(const float2*)(pqf + 2 * (t0 + g * 4 + 0));
        const float c = fmaf(pq.x, ret, pq.y * v4.x);
        ret = r4.x + (d4.x ? 0.0f : c); o.x = ret; }
      *(float4*)(outRow + t0 + g * 4) = o;        // b128 stores, L2 merges lines
    }
  }
}

// ---------------------------------------------------------------------------
extern "C" void kernel_launch(void* const* d_in, const int* in_sizes, int n_in,
                              void* d_out, int out_size, void* d_ws, size_t ws_size,
                              hipStream_t stream) {
  (void)n_in; (void)out_size; (void)d_ws; (void)ws_size;
  const float* raw_gamma = (const float*)d_in[0];
  const float* raw_lambd = (const float*)d_in[1];
  const float* values    = (const float*)d_in[2];
  const float* rewards   = (const float*)d_in[3];
  const int*   dones     = (const int*)d_in[4];

  const int B = in_sizes[3] / S_LEN;               // 8192
  float* out = (float*)d_out;
  float* srw = out + (size_t)B * S_LEN;            // second tuple element

  td_weights_kernel<<<1, 1024, 0, stream>>>(raw_gamma, raw_lambd, srw);

  const int rows_per_block = WAVES * 32;           // 256
  dim3 grid((unsigned)((B + rows_per_block - 1) / rows_per_block));
  td_scan_kernel<<<grid, BLOCK, SMEM_BYTES, stream>>>(
      raw_gamma, raw_lambd, values, rewards, dones, out);
}